// PinholeCamera_43482248905255
// MI455X (gfx1250) — compile-verified
//
#include <hip/hip_runtime.h>
#include <math.h>

#define THREADS 256
#define RAY_BLOCKS 2048
#define MAX_CAM_LDS 128   // 128 cams * 16 floats = 8 KB LDS

// ---------------------------------------------------------------------------
// Phase 1: per-camera pose refinement chain.
// cam_table[c*16] = { R row-major (9), t (3), fx, fy, cx, cy }
// where [R|t] = pose_invert(pose_compose(exp(se3_refine[c]), pose_invert(c2w[c])))
// ---------------------------------------------------------------------------
__global__ __launch_bounds__(128)
void pose_setup_kernel(const float* __restrict__ ks,
                       const float* __restrict__ c2ws,
                       const float* __restrict__ se3,
                       float* __restrict__ cam_table,
                       int n_cam)
{
    int c = blockIdx.x * blockDim.x + threadIdx.x;
    if (c >= n_cam) return;

    float P[12];
#pragma unroll
    for (int i = 0; i < 12; ++i) P[i] = c2ws[c * 12 + i];

    // rts0 = invert(c2w):  R0t = R^T,  t0 = -R^T * t
    float R0t[9] = { P[0], P[4], P[8],
                     P[1], P[5], P[9],
                     P[2], P[6], P[10] };
    float tv0 = P[3], tv1 = P[7], tv2 = P[11];
    float t0[3];
#pragma unroll
    for (int i = 0; i < 3; ++i)
        t0[i] = -(R0t[i*3+0]*tv0 + R0t[i*3+1]*tv1 + R0t[i*3+2]*tv2);

    // refine = se3_to_SE3(se3_refine[c])
    float w0 = se3[c*6+0], w1 = se3[c*6+1], w2 = se3[c*6+2];
    float u0 = se3[c*6+3], u1 = se3[c*6+4], u2 = se3[c*6+5];
    float wx[9] = { 0.f, -w2,  w1,
                    w2,  0.f, -w0,
                   -w1,  w0,  0.f };
    float wx2[9];
#pragma unroll
    for (int i = 0; i < 3; ++i)
#pragma unroll
        for (int j = 0; j < 3; ++j)
            wx2[i*3+j] = wx[i*3+0]*wx[0*3+j] + wx[i*3+1]*wx[1*3+j] + wx[i*3+2]*wx[2*3+j];

    float t2 = w0*w0 + w1*w1 + w2*w2;
    float th = sqrtf(t2);
    float A, B, C;
    if (th < 1e-3f) {
        A = 1.f   - t2 * (1.f/6.f);
        B = 0.5f  - t2 * (1.f/24.f);
        C = (1.f/6.f) - t2 * (1.f/120.f);
    } else {
        float s = sinf(th), co = cosf(th);
        A = s / th;
        B = (1.f - co) / t2;
        C = (th - s) / (t2 * th);
    }

    float Rr[9], V[9];
#pragma unroll
    for (int i = 0; i < 9; ++i) {
        float id = (i == 0 || i == 4 || i == 8) ? 1.f : 0.f;
        Rr[i] = id + A * wx[i] + B * wx2[i];
        V[i]  = id + B * wx[i] + C * wx2[i];
    }
    float tr[3];
#pragma unroll
    for (int i = 0; i < 3; ++i)
        tr[i] = V[i*3+0]*u0 + V[i*3+1]*u1 + V[i*3+2]*u2;

    // rts = compose(refine, rts0):  R = R0t @ Rr,  t = t0 + R0t @ tr
    float Rm[9], tm[3];
#pragma unroll
    for (int i = 0; i < 3; ++i) {
#pragma unroll
        for (int j = 0; j < 3; ++j)
            Rm[i*3+j] = R0t[i*3+0]*Rr[0*3+j] + R0t[i*3+1]*Rr[1*3+j] + R0t[i*3+2]*Rr[2*3+j];
        tm[i] = t0[i] + R0t[i*3+0]*tr[0] + R0t[i*3+1]*tr[1] + R0t[i*3+2]*tr[2];
    }

    // c2w_final = invert(rts):  Rg = Rm^T,  tg = -Rg @ tm
    float Rg[9], tg[3];
#pragma unroll
    for (int i = 0; i < 3; ++i)
#pragma unroll
        for (int j = 0; j < 3; ++j)
            Rg[i*3+j] = Rm[j*3+i];
#pragma unroll
    for (int i = 0; i < 3; ++i)
        tg[i] = -(Rg[i*3+0]*tm[0] + Rg[i*3+1]*tm[1] + Rg[i*3+2]*tm[2]);

    float* cb = cam_table + c * 16;
#pragma unroll
    for (int i = 0; i < 9; ++i) cb[i] = Rg[i];
    cb[9]  = tg[0]; cb[10] = tg[1]; cb[11] = tg[2];
    cb[12] = ks[c*9 + 0];   // fx
    cb[13] = ks[c*9 + 4];   // fy
    cb[14] = ks[c*9 + 2];   // cx
    cb[15] = ks[c*9 + 5];   // cy
}

// ---------------------------------------------------------------------------
// Phase 2: per-ray generation (HBM-bandwidth-bound streaming kernel).
// Camera table broadcast into LDS via CDNA5 async global->LDS copy; index
// stream loaded NT, output stream stored NT to keep L2 for the live data.
// ---------------------------------------------------------------------------
__global__ __launch_bounds__(THREADS)
void ray_gen_kernel(const float* __restrict__ cam_table,
                    const int* __restrict__ view_idx,
                    const int* __restrict__ ray_idx,
                    const int* __restrict__ Wp,
                    float* __restrict__ out,
                    int n_rays, int n_cam)
{
    __shared__ float cam_lds[MAX_CAM_LDS * 16];

    unsigned tid = threadIdx.x;
    unsigned nc = (unsigned)(n_cam > MAX_CAM_LDS ? MAX_CAM_LDS : n_cam);
    unsigned bytes = nc * 64u;  // 16 floats per camera
    unsigned lds_base = (unsigned)(size_t)(void*)&cam_lds[0];

    // Async DMA-style fill: each active lane copies 16B global -> LDS.
    for (unsigned off = tid * 16u; off < bytes; off += THREADS * 16u) {
        unsigned lds_addr = lds_base + off;
        asm volatile("global_load_async_to_lds_b128 %0, %1, %2"
                     :: "v"(lds_addr), "v"(off), "s"(cam_table)
                     : "memory");
    }
    asm volatile("s_wait_asynccnt 0x0" ::: "memory");
    __syncthreads();

    const int W = Wp[0];
    const float rW = 1.0f / (float)W;       // hoisted reciprocal
    const int stride = (int)(gridDim.x * blockDim.x);
    float* __restrict__ out_d = out + (size_t)n_rays * 3;
    float* __restrict__ out_u = out + (size_t)n_rays * 6;

    for (int i = (int)(blockIdx.x * blockDim.x + tid); i < n_rays; i += stride) {
        if (i + stride < n_rays) {
            __builtin_prefetch(&view_idx[i + stride], 0, 0);
            __builtin_prefetch(&ray_idx[i + stride], 0, 0);
        }
        int cam = __builtin_nontemporal_load(&view_idx[i]);
        int r   = __builtin_nontemporal_load(&ray_idx[i]);

        // y = r / W via reciprocal multiply + exact fixup (r < 2^21 fits f32)
        int y = (int)((float)r * rW);
        int x = r - y * W;
        if (x < 0)       { --y; x += W; }
        else if (x >= W) { ++y; x -= W; }

        const float* cb = &cam_lds[cam << 4];
        float fx = cb[12], fy = cb[13], cx = cb[14], cy = cb[15];
        float dx = ((float)x + 0.5f - cx) / fx;
        float dy = ((float)y + 0.5f - cy) / fy;

        float d0 = cb[0]*dx + cb[1]*dy + cb[2];
        float d1 = cb[3]*dx + cb[4]*dy + cb[5];
        float d2 = cb[6]*dx + cb[7]*dy + cb[8];
        float inv = 1.0f / sqrtf(d0*d0 + d1*d1 + d2*d2);

        float* po = out   + (size_t)i * 3;
        float* pd = out_d + (size_t)i * 3;
        float* pu = out_u + (size_t)i * 3;
        // NT scalar stores; LLVM merges adjacent ones into global_store_b96 NT
        __builtin_nontemporal_store(cb[9],   po + 0);
        __builtin_nontemporal_store(cb[10],  po + 1);
        __builtin_nontemporal_store(cb[11],  po + 2);
        __builtin_nontemporal_store(d0*inv,  pd + 0);
        __builtin_nontemporal_store(d1*inv,  pd + 1);
        __builtin_nontemporal_store(d2*inv,  pd + 2);
        __builtin_nontemporal_store(-cb[1],  pu + 0);
        __builtin_nontemporal_store(-cb[4],  pu + 1);
        __builtin_nontemporal_store(-cb[7],  pu + 2);
    }
}

// ---------------------------------------------------------------------------
extern "C" void kernel_launch(void* const* d_in, const int* in_sizes, int n_in,
                              void* d_out, int out_size, void* d_ws, size_t ws_size,
                              hipStream_t stream)
{
    const float* ks       = (const float*)d_in[0];
    const float* c2ws     = (const float*)d_in[1];
    const float* se3      = (const float*)d_in[2];
    const int*   view_idx = (const int*)d_in[3];
    const int*   ray_idx  = (const int*)d_in[4];
    // d_in[5] = H (unused; cy already encodes it)
    const int*   Wp       = (const int*)d_in[6];
    float*       out      = (float*)d_out;
    float*       cam_table = (float*)d_ws;   // n_cam * 16 floats

    int n_cam  = in_sizes[0] / 9;
    int n_rays = in_sizes[3];

    pose_setup_kernel<<<(n_cam + 127) / 128, 128, 0, stream>>>(
        ks, c2ws, se3, cam_table, n_cam);

    ray_gen_kernel<<<RAY_BLOCKS, THREADS, 0, stream>>>(
        cam_table, view_idx, ray_idx, Wp, out, n_rays, n_cam);
}